// SingleScaleNCC_730144441168
// MI455X (gfx1250) — compile-verified
//
#include <hip/hip_runtime.h>
#include <math.h>

// Problem constants (reference: shape (1,1,160,192,224), window_size = 9).
#define D_DIM 160
#define H_DIM 192
#define W_DIM 224
#define HALO  4          // win//2 for win=9

// Output tile per block and padded input tile.
#define OTD 8
#define OTH 16
#define OTW 16
#define PTD 16           // OTD + 2*HALO
#define PTH 24           // OTH + 2*HALO
#define PTW 24           // OTW + 2*HALO
#define PSTRIDE 25       // padded inner stride for input tiles (bank-conflict free)
#define F1S 17           // inner stride of W-smoothed buffer

typedef float v2f __attribute__((ext_vector_type(2)));
typedef float v8f __attribute__((ext_vector_type(8)));

// ---- LDS layout (floats) ----
#define OFF_TI   0
#define SZ_TILE  (PTD*PTH*PSTRIDE)          // 9600
#define OFF_TJ   (OFF_TI + SZ_TILE)
#define OFF_F1   (OFF_TJ + SZ_TILE)
#define SZ_F1    (PTD*PTH*F1S)              // 6528
#define OFF_F2   (OFF_F1 + SZ_F1)
#define SZ_F2    (PTD*OTH*OTW)              // 4096
#define OFF_OUT  (OFF_F2 + SZ_F2)
#define SZ_OUT1  (OTD*OTH*OTW)              // 2048
#define OFF_WSUM (OFF_OUT + 5*SZ_OUT1)
#define SMEM_FLOATS (OFF_WSUM + 16)         // 40080 floats = 160320 bytes

static __device__ __forceinline__ float band9(int k, int m) {
    int d = k - m;
    return (d >= 0 && d <= 8) ? 1.0f : 0.0f;
}

// D = A(16x4) * B(4x16) + C, f32 WMMA (CDNA5, wave32).
static __device__ __forceinline__ v8f wmma_f32_16x16x4(v2f a, v2f b, v8f c) {
    return __builtin_amdgcn_wmma_f32_16x16x4_f32(
        /*neg_a=*/false, a, /*neg_b=*/false, b,
        /*c_mod=*/(short)0, c, /*reuse_a=*/false, /*reuse_b=*/false);
}

__global__ void __launch_bounds__(512, 1)
ncc_fused_kernel(const float* __restrict__ pred,   // Ji
                 const float* __restrict__ targ,   // Ii
                 float* __restrict__ partials) {
    extern __shared__ float smem[];
    float* tI   = smem + OFF_TI;
    float* tJ   = smem + OFF_TJ;
    float* F1   = smem + OFF_F1;
    float* F2   = smem + OFF_F2;
    float* OUT5 = smem + OFF_OUT;
    float* wsum = smem + OFF_WSUM;

    const int t    = threadIdx.x;       // 0..511
    const int lane = t & 31;
    const int wave = t >> 5;            // 0..15
    const int m16  = lane & 15;         // row (M) / column (N) index within wave halves
    const int kb   = (lane >> 4) << 1;  // K-pair base: lanes 0-15 -> K{0,1}, 16-31 -> K{2,3}

    // ---- Stage padded input tiles (zero padding at volume borders) ----
    const int base_z = (int)blockIdx.z * OTD - HALO;
    const int base_y = (int)blockIdx.y * OTH - HALO;
    const int base_x = (int)blockIdx.x * OTW - HALO;
    for (int i = t; i < PTD * PTH * PTW; i += blockDim.x) {
        int dz = i / (PTH * PTW);
        int r  = i % (PTH * PTW);
        int dy = r / PTW;
        int dx = r % PTW;
        int gz = base_z + dz, gy = base_y + dy, gx = base_x + dx;
        bool ok = (gz >= 0) & (gz < D_DIM) & (gy >= 0) & (gy < H_DIM) &
                  (gx >= 0) & (gx < W_DIM);
        long gi = ((long)gz * H_DIM + gy) * W_DIM + gx;
        float vi = ok ? targ[gi] : 0.0f;
        float vj = ok ? pred[gi] : 0.0f;
        int li = (dz * PTH + dy) * PSTRIDE + dx;
        tI[li] = vi;
        tJ[li] = vj;
    }

    // ---- Banded "ones" A-matrix (16x24 band, 6 K-chunks of 4), constant per lane ----
    v2f A[6];
#pragma unroll
    for (int c = 0; c < 6; ++c) {
        int k0 = 4 * c + kb;
        A[c].x = band9(k0,     m16);
        A[c].y = band9(k0 + 1, m16);
    }

    // ---- 5 fields: I, J, I*I, J*J, I*J -> separable 9^3 box sums via WMMA ----
#pragma unroll
    for (int f = 0; f < 5; ++f) {
        __syncthreads();

        // W-pass: F1[dz][dy][wx] = sum_{k=0..8} P[dz][dy][wx+k]; 24 line-groups.
        for (int g = wave; g < (PTD * PTH) / 16; g += 16) {
            int qn = g * 16 + m16;          // this lane's line (B column)
            int dz = qn / PTH;
            int dy = qn % PTH;
            int pb = (dz * PTH + dy) * PSTRIDE;
            v8f acc = {0.f, 0.f, 0.f, 0.f, 0.f, 0.f, 0.f, 0.f};
#pragma unroll
            for (int c = 0; c < 6; ++c) {
                int k0 = 4 * c + kb;
                float a0 = tI[pb + k0],     b0 = tJ[pb + k0];
                float a1 = tI[pb + k0 + 1], b1 = tJ[pb + k0 + 1];
                v2f B;
                B.x = (f == 0) ? a0 : (f == 1) ? b0 : (f == 2) ? a0 * a0
                                    : (f == 3) ? b0 * b0 : a0 * b0;
                B.y = (f == 0) ? a1 : (f == 1) ? b1 : (f == 2) ? a1 * a1
                                    : (f == 3) ? b1 * b1 : a1 * b1;
                acc = wmma_f32_16x16x4(A[c], B, acc);
            }
            int hi = lane >> 4;
#pragma unroll
            for (int v = 0; v < 8; ++v) {
                int wx = v + 8 * hi;        // D-matrix row M for this lane/VGPR
                F1[(dz * PTH + dy) * F1S + wx] = acc[v];
            }
        }
        __syncthreads();

        // H-pass: F2[dz][hy][wx] = sum_k F1[dz][hy+k][wx]; 16 groups (dz = wave).
        {
            int dz = wave;
            int wx = m16;
            v8f acc = {0.f, 0.f, 0.f, 0.f, 0.f, 0.f, 0.f, 0.f};
#pragma unroll
            for (int c = 0; c < 6; ++c) {
                int k0 = 4 * c + kb;
                v2f B;
                B.x = F1[(dz * PTH + k0) * F1S + wx];
                B.y = F1[(dz * PTH + k0 + 1) * F1S + wx];
                acc = wmma_f32_16x16x4(A[c], B, acc);
            }
            int hi = lane >> 4;
#pragma unroll
            for (int v = 0; v < 8; ++v) {
                int hy = v + 8 * hi;
                F2[(dz * OTH + hy) * OTW + wx] = acc[v];
            }
        }
        __syncthreads();

        // D-pass: OUT[od][hy][wx] = sum_k F2[od+k][hy][wx]; od in 0..7 valid
        // (rows 8..15 of the WMMA result are partial sums and are discarded).
        {
            int hy = wave;
            int wx = m16;
            v8f acc = {0.f, 0.f, 0.f, 0.f, 0.f, 0.f, 0.f, 0.f};
#pragma unroll
            for (int c = 0; c < 4; ++c) {   // K = 16 padded D positions
                int k0 = 4 * c + kb;
                v2f B;
                B.x = F2[(k0 * OTH + hy) * OTW + wx];
                B.y = F2[((k0 + 1) * OTH + hy) * OTW + wx];
                acc = wmma_f32_16x16x4(A[c], B, acc);
            }
            if (lane < 16) {                // rows 0..7 live in lanes 0..15
#pragma unroll
                for (int v = 0; v < 8; ++v) {
                    OUT5[f * SZ_OUT1 + (v * OTH + hy) * OTW + wx] = acc[v];
                }
            }
        }
    }
    __syncthreads();

    // ---- Combine: cc = sigma12^2 / max(sigma1*sigma2, eps); block partial sum ----
    const float inv  = 1.0f / 729.0f;         // 1/win^3, win=9
    const float eps  = 1.1920929e-07f;        // float32 machine eps
    float s = 0.0f;
    for (int i = t; i < SZ_OUT1; i += blockDim.x) {
        float bI  = OUT5[0 * SZ_OUT1 + i] * inv;   // mu1 (target)
        float bJ  = OUT5[1 * SZ_OUT1 + i] * inv;   // mu2 (pred)
        float bII = OUT5[2 * SZ_OUT1 + i] * inv;
        float bJJ = OUT5[3 * SZ_OUT1 + i] * inv;
        float bIJ = OUT5[4 * SZ_OUT1 + i] * inv;
        float s12 = bIJ - bI * bJ;
        float s11 = bII - bI * bI;
        float s22 = bJJ - bJ * bJ;
        float den = fmaxf(s11 * s22, eps);
        s += (s12 * s12) / den;
    }
    // Deterministic wave32 + cross-wave reduction (no atomics).
#pragma unroll
    for (int o = 16; o > 0; o >>= 1) s += __shfl_down(s, o, 32);
    if (lane == 0) wsum[wave] = s;
    __syncthreads();
    if (wave == 0) {
        float v = (lane < 16) ? wsum[lane] : 0.0f;
#pragma unroll
        for (int o = 16; o > 0; o >>= 1) v += __shfl_down(v, o, 32);
        if (lane == 0) {
            int bid = ((int)blockIdx.z * gridDim.y + blockIdx.y) * gridDim.x
                      + blockIdx.x;
            partials[bid] = v;
        }
    }
}

__global__ void __launch_bounds__(256)
ncc_finalize_kernel(const float* __restrict__ partials, int n,
                    float* __restrict__ out) {
    __shared__ float sh[256];
    float s = 0.0f;
    for (int i = threadIdx.x; i < n; i += blockDim.x) s += partials[i];
    sh[threadIdx.x] = s;
    __syncthreads();
#pragma unroll
    for (int o = 128; o > 0; o >>= 1) {
        if ((int)threadIdx.x < o) sh[threadIdx.x] += sh[threadIdx.x + o];
        __syncthreads();
    }
    if (threadIdx.x == 0)
        out[0] = -sh[0] / (float)((long)D_DIM * H_DIM * W_DIM);
}

extern "C" void kernel_launch(void* const* d_in, const int* in_sizes, int n_in,
                              void* d_out, int out_size, void* d_ws, size_t ws_size,
                              hipStream_t stream) {
    (void)in_sizes; (void)n_in; (void)out_size; (void)ws_size;
    // setup_inputs order: pred, target, window_size(=9, hardcoded).
    const float* pred = (const float*)d_in[0];
    const float* targ = (const float*)d_in[1];
    float* partials = (float*)d_ws;           // 3360 floats of scratch

    dim3 grid(W_DIM / OTW, H_DIM / OTH, D_DIM / OTD);   // 14 x 12 x 20 = 3360
    const int nblocks = grid.x * grid.y * grid.z;
    const size_t smem = (size_t)SMEM_FLOATS * sizeof(float);   // 160320 B

    hipFuncSetAttribute((const void*)ncc_fused_kernel,
                        hipFuncAttributeMaxDynamicSharedMemorySize, (int)smem);
    ncc_fused_kernel<<<grid, 512, smem, stream>>>(pred, targ, partials);
    ncc_finalize_kernel<<<1, 256, 0, stream>>>(partials, nblocks, (float*)d_out);
}